// TransitionUp_65042984730711
// MI455X (gfx1250) — compile-verified
//
#include <hip/hip_runtime.h>
#include <hip/hip_bf16.h>

// Problem constants (match reference)
#define BB    4
#define NN1   16384
#define NN2   4096
#define CIN   512
#define COUT  256
#define EPS_BN 1e-5f
#define EPS_D  1e-8f

typedef __attribute__((ext_vector_type(2))) float v2f;
typedef __attribute__((ext_vector_type(8))) float v8f;

// ---------------------------------------------------------------------------
// GEMM: Y[M x 256] = X[M x Kdim] * W[Kdim x 256]   (fp32, WMMA 16x16x4)
// One wave32 computes a 16(M) x 64(N) strip: 4 x (16x16) tiles sharing the
// A operand each k-step (4x less A traffic, 6.4 flop/byte vs 4).
// 8 waves (8 strips) per 256-thread block.
// A 16x4 layout (ISA 7.12.2): lane r (0..15) VGPR0=K0 VGPR1=K1, lanes 16..31
// hold K2/K3 for the same rows -> per-lane contiguous float2 at col k0+2*half.
// B 4x16: row striped across lanes within a VGPR, K rows split across halves.
// C/D: VGPR v, lanes 0..15 -> row m0+v, lanes 16..31 -> row m0+v+8.
// ---------------------------------------------------------------------------
__global__ __launch_bounds__(256) void gemm_wmma_f32(
    const float* __restrict__ X, const float* __restrict__ W,
    float* __restrict__ Y, int Kdim) {
  const int lane = threadIdx.x & 31;
  const int wave = threadIdx.x >> 5;
  const int r    = lane & 15;
  const int half = lane >> 4;

  const int wid = blockIdx.x * 8 + wave;     // global wave id == strip id
  const int tm  = wid >> 2;                  // 256/64 = 4 strips along N
  const int tg  = wid & 3;
  const int m0  = tm << 4;
  const int n0  = tg << 6;                   // 64-wide strip

  const float* __restrict__ arow = X + (size_t)(m0 + r) * Kdim + 2 * half;
  const float* __restrict__ bcol = W + (size_t)(2 * half) * COUT + n0 + r;

  v8f acc0 = {}, acc1 = {}, acc2 = {}, acc3 = {};
  for (int k0 = 0; k0 < Kdim; k0 += 4) {
    __builtin_prefetch(arow + k0 + 64, 0, 3);   // WGP-scope global_prefetch_b8
    v2f a;
    a.x = arow[k0];
    a.y = arow[k0 + 1];
    const float* __restrict__ bk = bcol + (size_t)k0 * COUT;

    v2f b0; b0.x = bk[0];  b0.y = bk[COUT + 0];
    v2f b1; b1.x = bk[16]; b1.y = bk[COUT + 16];
    v2f b2; b2.x = bk[32]; b2.y = bk[COUT + 32];
    v2f b3; b3.x = bk[48]; b3.y = bk[COUT + 48];

    acc0 = __builtin_amdgcn_wmma_f32_16x16x4_f32(false, a, false, b0, (short)0,
                                                 acc0, false, false);
    acc1 = __builtin_amdgcn_wmma_f32_16x16x4_f32(false, a, false, b1, (short)0,
                                                 acc1, false, false);
    acc2 = __builtin_amdgcn_wmma_f32_16x16x4_f32(false, a, false, b2, (short)0,
                                                 acc2, false, false);
    acc3 = __builtin_amdgcn_wmma_f32_16x16x4_f32(false, a, false, b3, (short)0,
                                                 acc3, false, false);
  }

  float* __restrict__ yrow = Y + (size_t)(m0 + 8 * half) * COUT + n0 + r;
#pragma unroll
  for (int v = 0; v < 8; ++v) {
    yrow[(size_t)v * COUT + 0]  = acc0[v];
    yrow[(size_t)v * COUT + 16] = acc1[v];
    yrow[(size_t)v * COUT + 32] = acc2[v];
    yrow[(size_t)v * COUT + 48] = acc3[v];
  }
}

// ---------------------------------------------------------------------------
// Zero the stats accumulators (4 x 256 floats).
// ---------------------------------------------------------------------------
__global__ void zero_stats(float* __restrict__ s) {
  s[threadIdx.x] = 0.0f;
}

// ---------------------------------------------------------------------------
// Column sums / sumsq over a [M x 256] matrix. One column per thread,
// rows_per_block rows per block, coalesced across threads, atomic combine.
// ---------------------------------------------------------------------------
__global__ __launch_bounds__(256) void col_stats(
    const float* __restrict__ Y, int rows_per_block,
    float* __restrict__ sum, float* __restrict__ sumsq) {
  const int t  = threadIdx.x;
  const size_t r0 = (size_t)blockIdx.x * rows_per_block;
  float s = 0.0f, s2 = 0.0f;
  for (int r = 0; r < rows_per_block; ++r) {
    float v = Y[(r0 + r) * COUT + t];
    s  += v;
    s2 += v * v;
  }
  atomicAdd(&sum[t], s);
  atomicAdd(&sumsq[t], s2);
}

// ---------------------------------------------------------------------------
// Per-column BN affine: a = g * rsqrt(var + eps), c = beta - a * mu.
// Handles both layers in one launch (256 threads).
// ---------------------------------------------------------------------------
__global__ void bn_finalize(
    const float* __restrict__ sum1, const float* __restrict__ sq1,
    const float* __restrict__ g1, const float* __restrict__ be1, float M1,
    const float* __restrict__ sum2, const float* __restrict__ sq2,
    const float* __restrict__ g2, const float* __restrict__ be2, float M2,
    float* __restrict__ a1, float* __restrict__ c1,
    float* __restrict__ a2, float* __restrict__ c2) {
  const int t = threadIdx.x;
  {
    float mu  = sum1[t] / M1;
    float var = sq1[t] / M1 - mu * mu;
    float a   = g1[t] * __frsqrt_rn(var + EPS_BN);
    a1[t] = a;
    c1[t] = be1[t] - a * mu;
  }
  {
    float mu  = sum2[t] / M2;
    float var = sq2[t] / M2 - mu * mu;
    float a   = g2[t] * __frsqrt_rn(var + EPS_BN);
    a2[t] = a;
    c2[t] = be2[t] - a * mu;
  }
}

// ---------------------------------------------------------------------------
// F2 = relu(a2 * Y2 + c2)   (materialized for the gather)
// ---------------------------------------------------------------------------
__global__ __launch_bounds__(256) void apply_bn_relu(
    const float* __restrict__ Y, const float* __restrict__ a,
    const float* __restrict__ c, float* __restrict__ F) {
  const int t = threadIdx.x;
  const size_t row = blockIdx.x;
  float v = a[t] * Y[row * COUT + t] + c[t];
  F[row * COUT + t] = fmaxf(v, 0.0f);
}

// ---------------------------------------------------------------------------
// Brute-force 3-NN per batch: all 4096 candidate points in LDS (48 KB),
// one query per thread, register top-3, inverse-distance weights.
// ---------------------------------------------------------------------------
__global__ __launch_bounds__(256) void knn3_kernel(
    const float* __restrict__ p1, const float* __restrict__ p2,
    int* __restrict__ idxb, float* __restrict__ wb) {
  __shared__ float spx[NN2];
  __shared__ float spy[NN2];
  __shared__ float spz[NN2];

  const int blocksPerBatch = NN1 / 256;              // 64
  const int b = blockIdx.x / blocksPerBatch;
  const int i = (blockIdx.x % blocksPerBatch) * 256 + threadIdx.x;

  const float* __restrict__ pb = p2 + (size_t)b * NN2 * 3;
  for (int j = threadIdx.x; j < NN2; j += 256) {
    spx[j] = pb[3 * j + 0];
    spy[j] = pb[3 * j + 1];
    spz[j] = pb[3 * j + 2];
  }
  __syncthreads();

  const int gq = b * NN1 + i;
  const float qx = p1[3 * gq + 0];
  const float qy = p1[3 * gq + 1];
  const float qz = p1[3 * gq + 2];

  float d0 = 3.4e38f, d1 = 3.4e38f, d2 = 3.4e38f;
  int   i0 = 0, i1 = 0, i2 = 0;

  for (int j = 0; j < NN2; ++j) {
    float dx = qx - spx[j];
    float dy = qy - spy[j];
    float dz = qz - spz[j];
    float d  = dx * dx + dy * dy + dz * dz;
    if (d < d2) {
      if (d < d1) {
        d2 = d1; i2 = i1;
        if (d < d0) { d1 = d0; i1 = i0; d0 = d; i0 = j; }
        else        { d1 = d;  i1 = j; }
      } else {
        d2 = d; i2 = j;
      }
    }
  }

  float r0 = 1.0f / (fmaxf(d0, 0.0f) + EPS_D);
  float r1 = 1.0f / (fmaxf(d1, 0.0f) + EPS_D);
  float r2 = 1.0f / (fmaxf(d2, 0.0f) + EPS_D);
  float inv = 1.0f / (r0 + r1 + r2);

  idxb[3 * gq + 0] = i0;
  idxb[3 * gq + 1] = i1;
  idxb[3 * gq + 2] = i2;
  wb[3 * gq + 0] = r0 * inv;
  wb[3 * gq + 1] = r1 * inv;
  wb[3 * gq + 2] = r2 * inv;
}

// ---------------------------------------------------------------------------
// out[q, :] = relu(a1*Y1[q,:]+c1) + sum_k w_k * F2[b*N2 + idx_k, :]
// Block = one query, threads = 256 channels -> fully coalesced gathers.
// ---------------------------------------------------------------------------
__global__ __launch_bounds__(256) void interp_kernel(
    const float* __restrict__ Y1, const float* __restrict__ a1,
    const float* __restrict__ c1, const float* __restrict__ F2,
    const int* __restrict__ idxb, const float* __restrict__ wb,
    float* __restrict__ out) {
  const int gq = blockIdx.x;
  const int t  = threadIdx.x;
  const int b  = gq / NN1;
  const size_t base = (size_t)b * NN2;

  const int   j0 = idxb[3 * gq + 0];
  const int   j1 = idxb[3 * gq + 1];
  const int   j2 = idxb[3 * gq + 2];
  const float w0 = wb[3 * gq + 0];
  const float w1 = wb[3 * gq + 1];
  const float w2 = wb[3 * gq + 2];

  float f1 = fmaxf(a1[t] * Y1[(size_t)gq * COUT + t] + c1[t], 0.0f);
  float g  = w0 * F2[(base + j0) * COUT + t]
           + w1 * F2[(base + j1) * COUT + t]
           + w2 * F2[(base + j2) * COUT + t];
  out[(size_t)gq * COUT + t] = f1 + g;
}

// ---------------------------------------------------------------------------
extern "C" void kernel_launch(void* const* d_in, const int* in_sizes, int n_in,
                              void* d_out, int out_size, void* d_ws, size_t ws_size,
                              hipStream_t stream) {
  const float* point_1 = (const float*)d_in[0];
  const float* feat_1  = (const float*)d_in[1];
  const float* point_2 = (const float*)d_in[2];
  const float* feat_2  = (const float*)d_in[3];
  const float* W1      = (const float*)d_in[4];
  // d_in[5] = b1 (cancels in BN)
  const float* g1      = (const float*)d_in[6];
  const float* be1     = (const float*)d_in[7];
  const float* W2      = (const float*)d_in[8];
  // d_in[9] = b2 (cancels in BN)
  const float* g2      = (const float*)d_in[10];
  const float* be2     = (const float*)d_in[11];
  float* out = (float*)d_out;

  const int M1 = BB * NN1;   // 65536
  const int M2 = BB * NN2;   // 16384

  char* ws = (char*)d_ws;
  float* Y1    = (float*)ws;  ws += (size_t)M1 * COUT * sizeof(float);   // 64 MB
  float* Y2    = (float*)ws;  ws += (size_t)M2 * COUT * sizeof(float);   // 16 MB
  float* F2    = (float*)ws;  ws += (size_t)M2 * COUT * sizeof(float);   // 16 MB
  float* stats = (float*)ws;  ws += 4 * COUT * sizeof(float);            // sum1,sq1,sum2,sq2
  float* coef  = (float*)ws;  ws += 4 * COUT * sizeof(float);            // a1,c1,a2,c2
  int*   idxb  = (int*)ws;    ws += (size_t)M1 * 3 * sizeof(int);        // 768 KB
  float* wb    = (float*)ws;                                             // 768 KB

  float* sum1 = stats;       float* sq1 = stats + 256;
  float* sum2 = stats + 512; float* sq2 = stats + 768;
  float* a1 = coef;        float* c1 = coef + 256;
  float* a2 = coef + 512;  float* c2 = coef + 768;

  // 1. zero BN accumulators
  zero_stats<<<1, 1024, 0, stream>>>(stats);

  // 2. GEMMs (WMMA fp32): strips = (M/16)*(256/64), 8 strips (waves) per block
  gemm_wmma_f32<<<(M1 / 16) * 4 / 8, 256, 0, stream>>>(feat_1, W1, Y1, COUT); // 2048 blocks
  gemm_wmma_f32<<<(M2 / 16) * 4 / 8, 256, 0, stream>>>(feat_2, W2, Y2, CIN);  // 512 blocks

  // 3. column stats (256 rows per block)
  col_stats<<<M1 / 256, 256, 0, stream>>>(Y1, 256, sum1, sq1);
  col_stats<<<M2 / 256, 256, 0, stream>>>(Y2, 256, sum2, sq2);

  // 4. BN affine coefficients
  bn_finalize<<<1, 256, 0, stream>>>(sum1, sq1, g1, be1, (float)M1,
                                     sum2, sq2, g2, be2, (float)M2,
                                     a1, c1, a2, c2);

  // 5. materialize f2 = relu(a2*Y2 + c2)
  apply_bn_relu<<<M2, 256, 0, stream>>>(Y2, a2, c2, F2);

  // 6. brute-force 3-NN + inverse-distance weights
  knn3_kernel<<<BB * (NN1 / 256), 256, 0, stream>>>(point_1, point_2, idxb, wb);

  // 7. fused relu(BN(Y1)) + weighted gather of F2
  interp_kernel<<<M1, 256, 0, stream>>>(Y1, a1, c1, F2, idxb, wb, out);
}